// hungarianMatcher_23974507446517
// MI455X (gfx1250) — compile-verified
//
#include <hip/hip_runtime.h>
#include <hip/hip_bf16.h>
#include <cstdint>

#define BB 4
#define NN 512
#define ALPHA_C 0.25f
#define EPS_C 1e-8f
#define INF_C 1e18f

typedef __attribute__((ext_vector_type(2))) float v2f;
typedef __attribute__((ext_vector_type(8))) float v8f;

// -------------------------------------------------------------------------
// Phase 1: cost matrix via WMMA outer-product + async global->LDS staging.
// Grid: BB * (NN/16) blocks of 256 threads (8 wave32s).
// Block (b, itile) computes C[b, itile*16 : itile*16+16, :].
// Each wave computes 4 consecutive 16x16 column tiles.
// -------------------------------------------------------------------------
__global__ __launch_bounds__(256) void hm_cost_kernel(
    const float* __restrict__ logits,   // (B,N)
    const float* __restrict__ pp,       // (B,N,2)
    const float* __restrict__ gp,       // (B,N,2)
    float* __restrict__ outC)           // (B,N,N)
{
    __shared__ __align__(16) float s_gt[2 * NN];   // gt points of batch b (4KB)
    __shared__ float s_cc[16];                     // cost_class for 16 rows

    const int tid = threadIdx.x;
    const int b   = blockIdx.x / (NN / 16);
    const int it  = blockIdx.x % (NN / 16);
    const int m0  = it * 16;

    // ---- CDNA5 async copy: 256 lanes x 16B = 4KB of gt points -> LDS ----
    {
        const float* gbase = gp + (size_t)b * NN * 2;
        unsigned goff  = (unsigned)(tid * 16);                       // byte offset
        unsigned laddr = (unsigned)(size_t)(&s_gt[tid * 4]);         // LDS byte addr (flat low 32b)
        asm volatile("global_load_async_to_lds_b128 %0, %1, %2"
                     :: "v"(laddr), "v"(goff), "s"((unsigned long long)(size_t)gbase)
                     : "memory");
    }

    // cost_class for the 16 rows of this tile (overlaps with async copy)
    if (tid < 16) {
        float x = logits[(size_t)b * NN + m0 + tid];
        float p = 1.0f / (1.0f + __expf(-x));
        float negc = (1.0f - ALPHA_C) * p * p * (-__logf(1.0f - p + EPS_C));
        float posc = ALPHA_C * (1.0f - p) * (1.0f - p) * (-__logf(p + EPS_C));
        s_cc[tid] = posc - negc;
    }

    asm volatile("s_wait_asynccnt 0" ::: "memory");   // this wave's async writes done
    __syncthreads();                                  // all waves' LDS visible

    const int wave = tid >> 5;
    const int lane = tid & 31;
    const int lr   = lane & 15;
    const bool lo  = (lane < 16);

    // A fragment (16x4 f32): lanes 0-15 hold row m0+lane, K0..1 = (p, -1);
    // lanes 16-31 (K2..3) are zero so K=4 contributes nothing extra.
    float pxv = pp[((size_t)b * NN + m0 + lr) * 2 + 0];
    float pyv = pp[((size_t)b * NN + m0 + lr) * 2 + 1];
    v2f ax, ay;
    ax.x = lo ? pxv : 0.0f;  ax.y = lo ? -1.0f : 0.0f;
    ay.x = lo ? pyv : 0.0f;  ay.y = lo ? -1.0f : 0.0f;

    v8f c0 = {};

    for (int t = 0; t < 4; ++t) {
        const int jt = wave * 4 + t;
        const int n0 = jt * 16;
        const int col = n0 + lr;
        float gxv = s_gt[2 * col + 0];
        float gyv = s_gt[2 * col + 1];
        // B fragment (4x16 f32): lanes 0-15 hold col n0+lane, K0..1 = (1, g)
        v2f bx, by;
        bx.x = lo ? 1.0f : 0.0f;  bx.y = lo ? gxv : 0.0f;
        by.x = lo ? 1.0f : 0.0f;  by.y = lo ? gyv : 0.0f;

        // dx[m][n] = px_m - gx_n ; dy[m][n] = py_m - gy_n  (matrix pipe)
        v8f dx = __builtin_amdgcn_wmma_f32_16x16x4_f32(
            false, ax, false, bx, (short)0, c0, false, false);
        v8f dy = __builtin_amdgcn_wmma_f32_16x16x4_f32(
            false, ay, false, by, (short)0, c0, false, false);

        // C/D layout: lanes 0-15 -> rows r, lanes 16-31 -> rows r+8; col = N=lane%16
        #pragma unroll
        for (int r = 0; r < 8; ++r) {
            int mloc = lo ? r : (r + 8);
            float c = s_cc[mloc] + fabsf(dx[r]) + fabsf(dy[r]);
            outC[((size_t)b * NN + (m0 + mloc)) * NN + col] = c;
        }
    }
}

// -------------------------------------------------------------------------
// Phase 2: assignment with the reference's *literal* semantics (mv/wy are
// never written back in the numpy code -> greedy chain with potentials,
// augmentation collapses to p[j0_final] = i). One block per batch,
// 512 threads = 16 wave32s, all state in LDS.
// -------------------------------------------------------------------------
__global__ __launch_bounds__(512) void hm_lsa_kernel(
    const float* __restrict__ C,     // (B,N,N)
    float* __restrict__ outPred,     // (B,N) as float
    float* __restrict__ outGt)       // (B,N) as float
{
    __shared__ float u[NN + 1];
    __shared__ float v[NN + 1];
    __shared__ int   p[NN + 1];
    __shared__ int   used[NN + 1];
    __shared__ float red_val[16];
    __shared__ int   red_idx[16];
    __shared__ int   s_j0;
    __shared__ float s_delta;
    __shared__ int   s_done;

    const int tid = threadIdx.x;           // 0..511
    const int b   = blockIdx.x;
    const float* Cb = C + (size_t)b * NN * NN;

    u[tid] = 0.0f; v[tid] = 0.0f; p[tid] = 0;
    if (tid == 0) { u[NN] = 0.0f; v[NN] = 0.0f; p[NN] = 0; }
    __syncthreads();

    for (int i = 1; i <= NN; ++i) {
        if (tid == 0) { p[0] = i; s_j0 = 0; s_done = 0; }
        used[tid] = 0;
        if (tid == 0) used[NN] = 0;
        __syncthreads();

        for (;;) {
            const int j0 = s_j0;
            if (tid == 0) used[j0] = 1;
            __syncthreads();

            const int i0 = p[j0];
            const int j  = tid + 1;               // column 1..N
            float val = INF_C;
            if (!used[j]) {
                val = Cb[(size_t)(i0 - 1) * NN + (j - 1)] - u[i0] - v[j];
            }
            int idx = j;

            // wave32 argmin (first-index tie-break, matches np.argmin)
            #pragma unroll
            for (int off = 16; off > 0; off >>= 1) {
                float ov = __shfl_down(val, off, 32);
                int   oi = __shfl_down(idx, off, 32);
                if (ov < val || (ov == val && oi < idx)) { val = ov; idx = oi; }
            }
            const int wv = tid >> 5;
            const int ln = tid & 31;
            if (ln == 0) { red_val[wv] = val; red_idx[wv] = idx; }
            __syncthreads();

            if (tid == 0) {
                float bv = red_val[0]; int bi = red_idx[0];
                #pragma unroll
                for (int w = 1; w < 16; ++w) {
                    float ov = red_val[w]; int oi = red_idx[w];
                    if (ov < bv || (ov == bv && oi < bi)) { bv = ov; bi = oi; }
                }
                s_delta = bv; s_j0 = bi;
                s_done = (p[bi] == 0) ? 1 : 0;
            }
            __syncthreads();

            const float delta = s_delta;
            const int   done  = s_done;
            // potentials: targets u[p[j]] are distinct over used j -> no atomics
            if (used[tid]) { u[p[tid]] += delta; v[tid] -= delta; }
            if (tid == 0 && used[NN]) { u[p[NN]] += delta; v[NN] -= delta; }
            __syncthreads();
            if (done) break;
        }
        if (tid == 0) p[s_j0] = i;    // literal augmentation (way[] stays 0)
        __syncthreads();
    }

    // pred_idx = arange; gt_idx: col_of_row[p[j]-1] = j-1
    outPred[(size_t)b * NN + tid] = (float)tid;
    const int r = p[tid + 1];
    outGt[(size_t)b * NN + (r - 1)] = (float)tid;
}

extern "C" void kernel_launch(void* const* d_in, const int* in_sizes, int n_in,
                              void* d_out, int out_size, void* d_ws, size_t ws_size,
                              hipStream_t stream) {
    const float* logits = (const float*)d_in[0];   // (B,N)
    const float* pp     = (const float*)d_in[1];   // (B,N,2)
    const float* gp     = (const float*)d_in[2];   // (B,N,2)
    // d_in[3] = gt_mask, all ones per reference assertion -> unused

    float* out     = (float*)d_out;
    float* outPred = out;                       // B*N
    float* outGt   = out + BB * NN;             // B*N
    float* outC    = out + 2 * BB * NN;         // B*N*N

    hm_cost_kernel<<<dim3(BB * (NN / 16)), dim3(256), 0, stream>>>(logits, pp, gp, outC);
    hm_lsa_kernel<<<dim3(BB), dim3(NN), 0, stream>>>(outC, outPred, outGt);
}